// GNN_graphpred_17961553232342
// MI455X (gfx1250) — compile-verified
//
#include <hip/hip_runtime.h>

#define D 128
#define XS_STRIDE (D + 4)   // LDS pad: stride 132 -> 16 lanes hit distinct banks
#define RT 5                // row tiles per block: 80 rows (50000 = 625 * 80)
#define ROWS (RT * 16)

typedef __attribute__((ext_vector_type(2))) float v2f;
typedef __attribute__((ext_vector_type(8))) float v8f;

// ---------------------------------------------------------------- zero fill
__global__ void zero_f32(float* __restrict__ p, int n) {
    int i = blockIdx.x * blockDim.x + threadIdx.x;
    if (i < n) p[i] = 0.0f;
}

// ------------------------------------------------- atom embedding: h[n] = sum_f atom_emb[f][x[n,f]]
// one wave32 per node, 4 features per lane
__global__ __launch_bounds__(256) void atom_embed(
    const int* __restrict__ x, const float* __restrict__ atom_emb,
    float* __restrict__ h, int N) {
    int wid  = (blockIdx.x * 256 + threadIdx.x) >> 5;
    int lane = threadIdx.x & 31;
    if (wid >= N) return;
    int f0 = lane * 4;
    float4 acc = make_float4(0.f, 0.f, 0.f, 0.f);
#pragma unroll
    for (int f = 0; f < 9; ++f) {
        int idx = x[wid * 9 + f];
        const float4 t = *(const float4*)(atom_emb + ((size_t)(f * 120 + idx)) * D + f0);
        acc.x += t.x; acc.y += t.y; acc.z += t.z; acc.w += t.w;
    }
    *(float4*)(h + (size_t)wid * D + f0) = acc;
}

// --------------------------------------- edge message + scatter: aggr[dst] += h[src] + bond_emb(ea)
// one wave32 per edge, 4 features per lane, f32 atomics (aggr is L2-resident: 25.6 MB << 192 MB)
__global__ __launch_bounds__(256) void edge_msg(
    const int* __restrict__ ei, const int* __restrict__ ea,
    const float* __restrict__ h, const float* __restrict__ bond_l,
    float* __restrict__ aggr, int E) {
    int wid  = (blockIdx.x * 256 + threadIdx.x) >> 5;
    int lane = threadIdx.x & 31;
    if (wid >= E) return;
    int src = ei[wid];
    int dst = ei[E + wid];
    int f0  = lane * 4;
    float4 m = *(const float4*)(h + (size_t)src * D + f0);
#pragma unroll
    for (int f = 0; f < 5; ++f) {
        int idx = ea[wid * 5 + f];
        const float4 t = *(const float4*)(bond_l + (size_t)(f * 7 + idx) * D + f0);
        m.x += t.x; m.y += t.y; m.z += t.z; m.w += t.w;
    }
    float* p = aggr + (size_t)dst * D + f0;
    atomicAdd(p + 0, m.x); atomicAdd(p + 1, m.y);
    atomicAdd(p + 2, m.z); atomicAdd(p + 3, m.w);
}

// ------------- z = (aggr + h + c_l) @ W^T + b, plus BN sum/sumsq accumulation
// c_l[k] = sum_f bond_emb[l][f][0][k] (self-loop edge embedding constant)
// block = 256 threads = 8 waves; block covers 80 rows; wave w owns cols [16w,16w+16)
// and computes 5 stacked 16x16 tiles, reusing each B fragment across all 5 A fragments.
__global__ __launch_bounds__(256) void gemm_bn_stats(
    const float* __restrict__ aggr, const float* __restrict__ h,
    const float* __restrict__ bond_l, const float* __restrict__ W,
    const float* __restrict__ bias, float* __restrict__ z,
    float* __restrict__ stats /* [0,D)=sum, [D,2D)=sumsq */) {
    __shared__ float csh[D];
    __shared__ float xs[ROWS * XS_STRIDE];   // 80*132*4 = 42.2 KB
    const int t = threadIdx.x;

    if (t < D) {
        float c = 0.f;
#pragma unroll
        for (int f = 0; f < 5; ++f) c += bond_l[f * 7 * D + t];
        csh[t] = c;
    }
    __syncthreads();

    const size_t row0 = (size_t)blockIdx.x * ROWS;
    // stage X tile (aggr + h + c) into LDS: 80 rows x 128 cols, 10 float4 per thread
    for (int i = t; i < ROWS * (D / 4); i += 256) {
        int rr = i >> 5;             // row in tile
        int cc = (i & 31) * 4;       // col
        size_t gi = (row0 + rr) * D + cc;
        float4 av = *(const float4*)(aggr + gi);
        float4 hv = *(const float4*)(h + gi);
        float* xp = xs + rr * XS_STRIDE + cc;
        xp[0] = av.x + hv.x + csh[cc + 0];
        xp[1] = av.y + hv.y + csh[cc + 1];
        xp[2] = av.z + hv.z + csh[cc + 2];
        xp[3] = av.w + hv.w + csh[cc + 3];
    }
    __syncthreads();

    const int wave = t >> 5;
    const int lane = t & 31;
    const int m    = lane & 15;              // A: row, B/C/D: col
    const int kh   = (lane >> 4) << 1;       // K sub-offset 0 or 2 (lane half)
    const int col0 = wave * 16;

    const float* __restrict__ xr = xs + m * XS_STRIDE;
    const float* __restrict__ wr = W + (size_t)(col0 + m) * D;   // B[k][n] = W[n][k]

    v8f acc[RT];
#pragma unroll
    for (int rt = 0; rt < RT; ++rt) acc[rt] = (v8f){};

#pragma unroll 2
    for (int K = 0; K < D; K += 4) {
        const int ka = K + kh;
        v2f b;
        b[0] = wr[ka];  b[1] = wr[ka + 1];
#pragma unroll
        for (int rt = 0; rt < RT; ++rt) {
            v2f a;
            a[0] = xr[rt * 16 * XS_STRIDE + ka];
            a[1] = xr[rt * 16 * XS_STRIDE + ka + 1];
            acc[rt] = __builtin_amdgcn_wmma_f32_16x16x4_f32(
                /*neg_a=*/false, a, /*neg_b=*/false, b,
                /*c_mod=*/(short)0, acc[rt], /*reuse_a=*/false, /*reuse_b=*/false);
        }
    }

    // C/D layout: VGPR r -> row r (lanes 0-15) / row r+8 (lanes 16-31), col = lane%16
    const int col   = col0 + m;
    const float bv  = bias[col];
    const int rbase = (lane >> 4) * 8;
    float s = 0.f, q = 0.f;
#pragma unroll
    for (int rt = 0; rt < RT; ++rt) {
#pragma unroll
        for (int r = 0; r < 8; ++r) {
            float val = acc[rt][r] + bv;
            z[(row0 + rt * 16 + rbase + r) * D + col] = val;
            s += val; q += val * val;
        }
    }
    // combine the two lane-halves sharing a column, then one atomic per column per wave
    s += __shfl_xor(s, 16, 32);
    q += __shfl_xor(q, 16, 32);
    if (lane < 16) {
        atomicAdd(&stats[col], s);
        atomicAdd(&stats[D + col], q);
    }
}

// ----------------------------------------- fold BN stats into per-feature scale/shift
__global__ void bn_finalize(const float* __restrict__ stats,
                            const float* __restrict__ gamma, const float* __restrict__ beta,
                            float* __restrict__ ab, float Ninv) {
    int c = threadIdx.x;   // 128 threads
    float mu  = stats[c] * Ninv;
    float var = stats[D + c] * Ninv - mu * mu;
    float inv = rsqrtf(var + 1e-5f);
    float a = gamma[c] * inv;
    ab[c]     = a;
    ab[D + c] = beta[c] - mu * a;
}

// ----------------------------------------- h = relu(z * a[col] + b[col])
__global__ void bn_relu_apply(const float* __restrict__ z, const float* __restrict__ ab,
                              float* __restrict__ h, int nvec /* N*D/4 */) {
    int i = blockIdx.x * blockDim.x + threadIdx.x;
    if (i >= nvec) return;
    int col = (i & 31) * 4;
    float4 v = *(const float4*)(z + (size_t)i * 4);
    float4 a = *(const float4*)(ab + col);
    float4 b = *(const float4*)(ab + D + col);
    v.x = fmaxf(fmaf(v.x, a.x, b.x), 0.f);
    v.y = fmaxf(fmaf(v.y, a.y, b.y), 0.f);
    v.z = fmaxf(fmaf(v.z, a.z, b.z), 0.f);
    v.w = fmaxf(fmaf(v.w, a.w, b.w), 0.f);
    *(float4*)(h + (size_t)i * 4) = v;
}

// ----------------------------------------- global mean pool (sum + count via atomics)
__global__ __launch_bounds__(256) void pool_kernel(
    const float* __restrict__ h, const int* __restrict__ batch,
    float* __restrict__ psum, float* __restrict__ pcnt, int N) {
    int wid  = (blockIdx.x * 256 + threadIdx.x) >> 5;
    int lane = threadIdx.x & 31;
    if (wid >= N) return;
    int g  = batch[wid];
    int f0 = lane * 4;
    const float4 v = *(const float4*)(h + (size_t)wid * D + f0);
    float* p = psum + (size_t)g * D + f0;
    atomicAdd(p + 0, v.x); atomicAdd(p + 1, v.y);
    atomicAdd(p + 2, v.z); atomicAdd(p + 3, v.w);
    if (lane == 0) atomicAdd(pcnt + g, 1.0f);
}

// ----------------------------------------- MLP head: one block per graph, 64 threads
__global__ void head_kernel(const float* __restrict__ psum, const float* __restrict__ pcnt,
                            const float* __restrict__ w1, const float* __restrict__ b1,
                            const float* __restrict__ w2, const float* __restrict__ b2,
                            float* __restrict__ out) {
    __shared__ float red[64];
    int g = blockIdx.x;
    int t = threadIdx.x;   // 64
    float inv = 1.0f / fmaxf(pcnt[g], 1.0f);
    float acc = 0.f;
    const float* gp = psum + (size_t)g * D;
    const float* wp = w1 + (size_t)t * D;
#pragma unroll 4
    for (int k = 0; k < D; ++k) acc = fmaf(gp[k] * inv, wp[k], acc);
    float hid = fmaxf(acc + b1[t], 0.f);
    red[t] = hid * w2[t];
    __syncthreads();
    for (int s = 32; s > 0; s >>= 1) {
        if (t < s) red[t] += red[t + s];
        __syncthreads();
    }
    if (t == 0) out[g] = red[0] + b2[0];
}

// =================================================================================
extern "C" void kernel_launch(void* const* d_in, const int* in_sizes, int n_in,
                              void* d_out, int out_size, void* d_ws, size_t ws_size,
                              hipStream_t stream) {
    const int*   x          = (const int*)d_in[0];
    const int*   edge_index = (const int*)d_in[1];
    const int*   edge_attr  = (const int*)d_in[2];
    const int*   batch      = (const int*)d_in[3];
    const float* atom_emb   = (const float*)d_in[4];
    const float* bond_emb   = (const float*)d_in[5];   // [L,5,7,D]
    const float* lin_w      = (const float*)d_in[6];   // [L,D,D]
    const float* lin_b      = (const float*)d_in[7];   // [L,D]
    const float* bn_gamma   = (const float*)d_in[8];
    const float* bn_beta    = (const float*)d_in[9];
    const float* mlp_w1     = (const float*)d_in[10];  // [64,128]
    const float* mlp_b1     = (const float*)d_in[11];
    const float* mlp_w2     = (const float*)d_in[12];  // [1,64]
    const float* mlp_b2     = (const float*)d_in[13];
    float* out = (float*)d_out;

    const int N  = in_sizes[3];        // 50000 (= 625 * 80)
    const int E  = in_sizes[2] / 5;    // 600000
    const int NG = 256;

    float* ws    = (float*)d_ws;
    float* h     = ws;
    float* aggr  = h    + (size_t)N * D;
    float* z     = aggr + (size_t)N * D;
    float* stats = z    + (size_t)N * D;        // 512 floats: sum, sumsq, a, b
    float* psum  = stats + 512;                 // NG*D
    float* pcnt  = psum + (size_t)NG * D;       // NG

    const int nwb = (N + 7) / 8;   // blocks for wave-per-node kernels (8 waves/block)

    atom_embed<<<nwb, 256, 0, stream>>>(x, atom_emb, h, N);

    for (int l = 0; l < 4; ++l) {
        const float* bond_l = bond_emb + (size_t)l * 5 * 7 * D;
        const float* Wl     = lin_w + (size_t)l * D * D;
        const float* bl     = lin_b + (size_t)l * D;

        zero_f32<<<(N * D + 255) / 256, 256, 0, stream>>>(aggr, N * D);
        zero_f32<<<1, 256, 0, stream>>>(stats, 256);
        edge_msg<<<(E + 7) / 8, 256, 0, stream>>>(edge_index, edge_attr, h, bond_l, aggr, E);
        gemm_bn_stats<<<N / ROWS, 256, 0, stream>>>(aggr, h, bond_l, Wl, bl, z, stats);
        bn_finalize<<<1, D, 0, stream>>>(stats, bn_gamma + l * D, bn_beta + l * D,
                                         stats + 256, 1.0f / (float)N);
        bn_relu_apply<<<(N * D / 4 + 255) / 256, 256, 0, stream>>>(z, stats + 256, h, N * D / 4);
    }

    zero_f32<<<(NG * D + NG + 255) / 256, 256, 0, stream>>>(psum, NG * D + NG);
    pool_kernel<<<nwb, 256, 0, stream>>>(h, batch, psum, pcnt, N);
    head_kernel<<<NG, 64, 0, stream>>>(psum, pcnt, mlp_w1, mlp_b1, mlp_w2, mlp_b2, out);
}